// FullyConnectedTensorProduct_36146444763337
// MI455X (gfx1250) — compile-verified
//
#include <hip/hip_runtime.h>

typedef __attribute__((ext_vector_type(2))) float v2f;
typedef __attribute__((ext_vector_type(8))) float v8f;

#define NSAMP   2048
#define DIMTOT  576
#define T1S     325   // LDS stride per A-row (64*5 + 5 pad -> stride 5 mod 64 banks)

// ---------------------------------------------------------------------------
// On-device Clebsch-Gordan generation (exact replica of the reference math,
// done in f64).  Real-basis CG is purely real for even l1+l2+l3 (all 11 paths).
// ---------------------------------------------------------------------------
__device__ __forceinline__ double dfact(int n) {
  double r = 1.0;
  for (int i = 2; i <= n; ++i) r *= (double)i;
  return r;
}

__device__ double su2_cg(int j1, int j2, int j3, int m1, int m2, int m3) {
  if (m3 != m1 + m2) return 0.0;
  double pre = sqrt((double)(2 * j3 + 1) * dfact(j3 + j1 - j2) * dfact(j3 - j1 + j2) *
                    dfact(j1 + j2 - j3) / dfact(j1 + j2 + j3 + 1));
  pre *= sqrt(dfact(j3 + m3) * dfact(j3 - m3) * dfact(j1 - m1) * dfact(j1 + m1) *
              dfact(j2 - m2) * dfact(j2 + m2));
  int k0 = 0;
  if (j2 - j3 - m1 > k0) k0 = j2 - j3 - m1;
  if (j1 - j3 + m2 > k0) k0 = j1 - j3 + m2;
  int k1 = j1 + j2 - j3;
  if (j1 - m1 < k1) k1 = j1 - m1;
  if (j2 + m2 < k1) k1 = j2 + m2;
  double s = 0.0;
  for (int k = k0; k <= k1; ++k) {
    double d = dfact(k) * dfact(j1 + j2 - j3 - k) * dfact(j1 - m1 - k) *
               dfact(j2 + m2 - k) * dfact(j3 - j2 + m1 + k) * dfact(j3 - j1 - m2 + k);
    s += ((k & 1) ? -1.0 : 1.0) / d;
  }
  return pre * s;
}

// Element (r,c) of the complex->real change-of-basis matrix for angular momentum l.
__device__ __forceinline__ void rb(int l, int r, int c, double& re, double& im) {
  re = 0.0; im = 0.0;
  if (r == l) { if (c == l) re = 1.0; return; }
  const double s2 = 0.70710678118654752440;
  int m = r - l;
  if (m > 0) {
    double sgn = (m & 1) ? -1.0 : 1.0;
    if (c == l + m) re = sgn * s2;
    else if (c == l - m) re = s2;
  } else {
    int mm = -m;
    double sgn = (mm & 1) ? -1.0 : 1.0;
    if (c == l + mm) im = -sgn * s2;
    else if (c == l - mm) im = s2;
  }
}

__global__ void cg_init(float* __restrict__ cg) {
  const int L1[11] = {0, 0, 0, 1, 1, 1, 1, 2, 2, 2, 2};
  const int L2[11] = {0, 1, 2, 0, 1, 1, 2, 0, 1, 2, 2};
  const int L3[11] = {0, 1, 2, 1, 0, 2, 1, 2, 1, 0, 2};
  int p = blockIdx.x;
  int l1 = L1[p], l2 = L2[p], l3 = L3[p];
  int d1 = 2 * l1 + 1, d2 = 2 * l2 + 1, d3 = 2 * l3 + 1;
  int t = threadIdx.x;
  if (t >= d1 * d2 * d3) return;
  int i = t / (d2 * d3);
  int j = (t / d3) % d2;
  int k = t % d3;
  double cr = 0.0;
  for (int a = 0; a < d1; ++a) {
    double a1r, a1i; rb(l1, i, a, a1r, a1i);
    if (a1r == 0.0 && a1i == 0.0) continue;
    for (int b = 0; b < d2; ++b) {
      double a2r, a2i; rb(l2, j, b, a2r, a2i);
      if (a2r == 0.0 && a2i == 0.0) continue;
      double tr = a1r * a2r - a1i * a2i;
      double ti = a1r * a2i + a1i * a2r;
      for (int c = 0; c < d3; ++c) {
        double g = su2_cg(l1, l2, l3, a - l1, b - l2, c - l3);
        if (g == 0.0) continue;
        double a3r, a3i; rb(l3, k, c, a3r, a3i);
        cr += g * (tr * a3r + ti * a3i);   // real( A1 * A2 * conj(A3) * cg )
      }
    }
  }
  cg[p * 125 + (i * 5 + j) * 5 + k] = (float)cr;
}

// ---------------------------------------------------------------------------
// Main fused tensor-product GEMM
// ---------------------------------------------------------------------------
struct Args {
  const float* x1;
  const float* x2;
  const float* w[11];
  const float* cg;
  float* out;
};

__device__ __forceinline__ constexpr int loff(int l) { return l == 0 ? 0 : (l == 1 ? 64 : 256); }

template <int L3, int L1p, int L2p, int WI>
__device__ __forceinline__ void do_path(const Args& args, float* t1s,
                                        int lane, int mlane, int half,
                                        int rowbase, int nL,
                                        v8f& acc0, v8f& acc1, v8f& acc2, v8f& acc3) {
  constexpr int d1 = 2 * L1p + 1;
  constexpr int d2 = 2 * L2p + 1;
  constexpr int d3 = 2 * L3 + 1;
  constexpr int off1 = loff(L1p);
  constexpr int off2 = loff(L2p);

  const float* __restrict__ x1 = args.x1;
  const float* __restrict__ x2 = args.x2;
  const float* __restrict__ Wp = args.w[WI];
  const float* __restrict__ cgp = args.cg + WI * 125;

  // ---- stage t1[m][v][i] = sum_j CG[i,j,k(m)] * x2[n(m), v, j] into LDS ----
  __syncthreads();
  for (int pair = lane; pair < 16 * 64; pair += 32) {
    int m = pair >> 6;
    int v = pair & 63;
    int row = rowbase + m;
    int n = row / d3;
    int k = row - n * d3;
    float c2v[d2];
#pragma unroll
    for (int j = 0; j < d2; ++j) c2v[j] = x2[n * DIMTOT + off2 + v * d2 + j];
#pragma unroll
    for (int i = 0; i < d1; ++i) {
      float s = 0.f;
#pragma unroll
      for (int j = 0; j < d2; ++j) s += cgp[(i * 5 + j) * 5 + k] * c2v[j];
      t1s[m * T1S + v * 5 + i] = s;
    }
  }
  __syncthreads();

  // ---- K loop: K = u*64 + v, A built on the fly, B streamed from L2 ----
  const float* wb = Wp + (half * 2) * 64 + mlane;  // B lane base: (K = 2*half + vgpr, N = mlane)
  const int mBase = mlane * T1S;
  const float* c1base = x1 + nL * DIMTOT + off1;

  for (int u = 0; u < 64; ++u) {
    float c1r[d1];
#pragma unroll
    for (int i = 0; i < d1; ++i) c1r[i] = c1base[u * d1 + i];

#pragma unroll
    for (int vb = 0; vb < 16; ++vb) {
      const int v0 = vb * 4 + half * 2;
      float ax = 0.f, ay = 0.f;
#pragma unroll
      for (int i = 0; i < d1; ++i) {
        ax += c1r[i] * t1s[mBase + v0 * 5 + i];
        ay += c1r[i] * t1s[mBase + v0 * 5 + 5 + i];
      }
      v2f a = {ax, ay};

      const float* wk = wb + (u * 64 + vb * 4) * 64;
      v2f b0 = {wk[0],  wk[64]};
      v2f b1 = {wk[16], wk[80]};
      v2f b2 = {wk[32], wk[96]};
      v2f b3 = {wk[48], wk[112]};

      acc0 = __builtin_amdgcn_wmma_f32_16x16x4_f32(false, a, false, b0, (short)0, acc0, false, false);
      acc1 = __builtin_amdgcn_wmma_f32_16x16x4_f32(false, a, false, b1, (short)0, acc1, false, false);
      acc2 = __builtin_amdgcn_wmma_f32_16x16x4_f32(false, a, false, b2, (short)0, acc2, false, false);
      acc3 = __builtin_amdgcn_wmma_f32_16x16x4_f32(false, a, false, b3, (short)0, acc3, false, false);
    }
  }
}

template <int L3>
__launch_bounds__(32)
__global__ void tp_kernel(Args args) {
  constexpr int d3 = 2 * L3 + 1;
  constexpr int off3 = loff(L3);
  __shared__ float t1s[16 * T1S];

  const int lane = threadIdx.x;
  const int mlane = lane & 15;
  const int half = lane >> 4;
  const int rowbase = blockIdx.x * 16;
  const int nL = (rowbase + mlane) / d3;  // sample index for this lane's A row

  v8f acc0 = {}, acc1 = {}, acc2 = {}, acc3 = {};

  if constexpr (L3 == 0) {
    do_path<0, 0, 0, 0>(args, t1s, lane, mlane, half, rowbase, nL, acc0, acc1, acc2, acc3);
    do_path<0, 1, 1, 4>(args, t1s, lane, mlane, half, rowbase, nL, acc0, acc1, acc2, acc3);
    do_path<0, 2, 2, 9>(args, t1s, lane, mlane, half, rowbase, nL, acc0, acc1, acc2, acc3);
  } else if constexpr (L3 == 1) {
    do_path<1, 0, 1, 1>(args, t1s, lane, mlane, half, rowbase, nL, acc0, acc1, acc2, acc3);
    do_path<1, 1, 0, 3>(args, t1s, lane, mlane, half, rowbase, nL, acc0, acc1, acc2, acc3);
    do_path<1, 1, 2, 6>(args, t1s, lane, mlane, half, rowbase, nL, acc0, acc1, acc2, acc3);
    do_path<1, 2, 1, 8>(args, t1s, lane, mlane, half, rowbase, nL, acc0, acc1, acc2, acc3);
  } else {
    do_path<2, 0, 2, 2>(args, t1s, lane, mlane, half, rowbase, nL, acc0, acc1, acc2, acc3);
    do_path<2, 1, 1, 5>(args, t1s, lane, mlane, half, rowbase, nL, acc0, acc1, acc2, acc3);
    do_path<2, 2, 0, 7>(args, t1s, lane, mlane, half, rowbase, nL, acc0, acc1, acc2, acc3);
    do_path<2, 2, 2, 10>(args, t1s, lane, mlane, half, rowbase, nL, acc0, acc1, acc2, acc3);
  }

  // D layout: lane (mlane, half), vgpr r -> element (M = r + 8*half, N = mlane)
#pragma unroll
  for (int r = 0; r < 8; ++r) {
    int row = rowbase + r + 8 * half;
    int n = row / d3;
    int k = row - n * d3;
    float* op = args.out + n * DIMTOT + off3 + mlane * d3 + k;  // w = mlane + 16*c
    op[0]          = acc0[r];
    op[16 * d3]    = acc1[r];
    op[32 * d3]    = acc2[r];
    op[48 * d3]    = acc3[r];
  }
}

// ---------------------------------------------------------------------------
extern "C" void kernel_launch(void* const* d_in, const int* in_sizes, int n_in,
                              void* d_out, int out_size, void* d_ws, size_t ws_size,
                              hipStream_t stream) {
  (void)in_sizes; (void)n_in; (void)out_size; (void)ws_size;

  Args args;
  args.x1 = (const float*)d_in[0];
  args.x2 = (const float*)d_in[1];
  for (int i = 0; i < 11; ++i) args.w[i] = (const float*)d_in[2 + i];
  args.cg = (const float*)d_ws;
  args.out = (float*)d_out;

  // Recompute CG tensors every call (deterministic, graph-capture safe, ~us).
  cg_init<<<11, 128, 0, stream>>>((float*)d_ws);

  // rows = N*(2*l3+1): 2048 / 6144 / 10240 -> 128 / 384 / 640 16-row tiles.
  tp_kernel<0><<<128, 32, 0, stream>>>(args);
  tp_kernel<1><<<384, 32, 0, stream>>>(args);
  tp_kernel<2><<<640, 32, 0, stream>>>(args);
}